// CrossFrameConsistencyLoss_58299886076510
// MI455X (gfx1250) — compile-verified
//
#include <hip/hip_runtime.h>

// CrossFrameConsistencyLoss for MI455X (gfx1250, wave32).
//
// Per-frame unique scene indices => invert to pos[f][p] (slot or -1), then one
// wave handles TWO scene points. A 16x128 matrix X (rows 0-7 = point A's 8
// frame vectors, rows 8-15 = point B's; absent frames read a zero dummy row)
// is fed to a chain of V_WMMA_F32_16X16X4_F32 as both A and B operands,
// producing D = X*X^T. The two 8x8 diagonal blocks are the Gram matrices; all
// loss terms (pair diff^2, x.s, ||s||^2, diag norms) are Gram entries.
//
// K-remap trick: Gram is invariant under feature-index permutation provided
// the index assignment per (lane-half, vgpr, chunk) is uniform across the 16
// lanes of a half. We assign half h -> features [64h, 64h+64), chunk c ->
// features 64h+2c, 64h+2c+1. Each lane then streams a contiguous 256B with
// 16 unconditional b128 loads (no exec-mask dance, schedulable clauses).
//
// Traffic: ~255 MB gathered feature rows read once (~12 us at 23.3 TB/s) +
// ~13 MB pos reads. WMMA work (6.6 GFLOP f32) is negligible.

#define F_FRAMES 8
#define N_PTS    65536
#define D_FEAT   128
#define P_SCENE  200000
#define N_PAIR   (P_SCENE / 2)
#define ACC_SLOTS 32
// acc slot map: [0..7] err_sum[f], [8..15] n_valid[f],
//               [16..22] pair_sum[fp], [24..30] n_common[fp]
// ws layout: [0,128)   float acc[32]
//            [128,640) float zrow[128]  (zero dummy feature row)
//            [1024, +6.4MB) int pos[F][P]

typedef float v2f __attribute__((ext_vector_type(2)));
typedef float v8f __attribute__((ext_vector_type(8)));

__global__ void cfc_init(float* __restrict__ wsf, int* __restrict__ pos, int total) {
  int i = blockIdx.x * blockDim.x + threadIdx.x;
  if (i < ACC_SLOTS + D_FEAT) wsf[i] = 0.0f;   // acc[32] + zrow[128]
  for (int k = i; k < total; k += gridDim.x * blockDim.x) pos[k] = -1;
}

__global__ void cfc_scatter(const int* __restrict__ idx, int* __restrict__ pos) {
  int i = blockIdx.x * blockDim.x + threadIdx.x;
  if (i >= F_FRAMES * N_PTS) return;
  int f = i >> 16;                 // N_PTS == 65536
  int p = idx[i];
  if (p >= 0) pos[f * P_SCENE + p] = i & (N_PTS - 1);
}

__device__ __forceinline__ float vsel8(v8f v, int j) {
  float r = v[0];
  r = (j == 1) ? v[1] : r;
  r = (j == 2) ? v[2] : r;
  r = (j == 3) ? v[3] : r;
  r = (j == 4) ? v[4] : r;
  r = (j == 5) ? v[5] : r;
  r = (j == 6) ? v[6] : r;
  r = (j == 7) ? v[7] : r;
  return r;
}

__global__ void __launch_bounds__(256)
cfc_main(const float* __restrict__ feat, const int* __restrict__ pos,
         const float* __restrict__ zrow, float* __restrict__ gacc) {
  __shared__ float s_acc[ACC_SLOTS];
  if (threadIdx.x < ACC_SLOTS) s_acc[threadIdx.x] = 0.0f;
  __syncthreads();

  const int lane  = threadIdx.x & 31;
  const int half  = lane >> 4;       // owns features [64*half, 64*half+64)
  const int row   = lane & 15;       // WMMA A-matrix row this lane feeds
  const int frame = row & 7;         // rows 0-7: point A, rows 8-15: point B
  const int j     = lane & 7;        // Gram column owned by result lanes

  const int nwaves = (gridDim.x * blockDim.x) >> 5;
  const int gwave  = (blockIdx.x * blockDim.x + threadIdx.x) >> 5;

  float errAcc = 0.f, nvAcc = 0.f, pairAcc = 0.f, ncAcc = 0.f;

  for (int t = gwave; t < N_PAIR; t += nwaves) {  // wave-uniform (EXEC all 1s at WMMA)
    const int pA = 2 * t, pB = 2 * t + 1;
    const int p  = (row < 8) ? pA : pB;
    const int pv = pos[frame * P_SCENE + p];
    const bool havev = (pv >= 0);

    const unsigned bal   = (unsigned)__ballot(havev);
    const unsigned maskA = bal & 0xFFu;
    const unsigned maskB = (bal >> 8) & 0xFFu;

    // Unconditional loads: absent rows read the shared zero row (L0-resident).
    const float* rp =
        (havev ? (feat + ((size_t)frame * N_PTS + (size_t)pv) * D_FEAT) : zrow)
        + 64 * half;

    v8f g0 = {}, g1 = {};  // dual accumulators halve the WMMA RAW chain
#pragma unroll
    for (int m = 0; m < 16; ++m) {
      const float4 q = *(const float4*)(rp + 4 * m);
      v2f a0; a0.x = q.x; a0.y = q.y;   // features 64h+4m, 64h+4m+1
      v2f a1; a1.x = q.z; a1.y = q.w;   // features 64h+4m+2, 64h+4m+3
      g0 = __builtin_amdgcn_wmma_f32_16x16x4_f32(false, a0, false, a0,
                                                 (short)0, g0, false, false);
      g1 = __builtin_amdgcn_wmma_f32_16x16x4_f32(false, a1, false, a1,
                                                 (short)0, g1, false, false);
    }
    const v8f g = g0 + g1;

    // Result layout: vgpr v, lanes 0-15 = D[v][lane]; lanes 16-31 = D[v+8][lane].
    // => lanes 0-7 own Gram_A column j (rows 0-7 in g[0..7]);
    //    lanes 24-31 own Gram_B column j=lane-24 (rows 0-7 of block B).
    float colsum = g[0] + g[1] + g[2] + g[3] + g[4] + g[5] + g[6] + g[7]; // x_j.s
    float diag   = vsel8(g, j);                 // G[j][j] = ||x_j||^2
    float offd   = vsel8(g, (j + 1) & 7);       // G[j+1][j] = G[j][j+1]
    float diagN  = __shfl_down(diag, 1, 32);    // G[j+1][j+1] from neighbor lane
    float ssq    = colsum;                      // ||s||^2 via closed 8-lane xor groups
    ssq += __shfl_xor(ssq, 1, 32);
    ssq += __shfl_xor(ssq, 2, 32);
    ssq += __shfl_xor(ssq, 4, 32);

    const bool isA = (lane < 8), isB = (lane >= 24);
    if (isA || isB) {
      const unsigned m8 = isA ? maskA : maskB;
      const int c = __popc(m8);
      if (c > 0 && ((m8 >> j) & 1u)) {
        const float inv = 1.0f / (float)c;
        errAcc += diag - 2.0f * colsum * inv + ssq * inv * inv; // ||x_j - mean||^2
        nvAcc  += 1.0f;
      }
      if (j < 7 && (((m8 >> j) & 3u) == 3u)) {
        pairAcc += diag + diagN - 2.0f * offd;                  // ||x_j - x_{j+1}||^2
        ncAcc   += 1.0f;
      }
    }
  }

  if (lane < 8 || lane >= 24) {
    atomicAdd(&s_acc[j], errAcc);
    atomicAdd(&s_acc[8 + j], nvAcc);
    if (j < 7) {
      atomicAdd(&s_acc[16 + j], pairAcc);
      atomicAdd(&s_acc[24 + j], ncAcc);
    }
  }
  __syncthreads();
  if (threadIdx.x < ACC_SLOTS) atomicAdd(&gacc[threadIdx.x], s_acc[threadIdx.x]);
}

__global__ void cfc_finalize(const float* __restrict__ acc, float* __restrict__ out) {
  if (blockIdx.x == 0 && threadIdx.x == 0) {
    float cons = 0.f;
    for (int fp = 0; fp < F_FRAMES - 1; ++fp) {
      float ps = acc[16 + fp], nc = acc[24 + fp];
      cons += (nc > 0.5f) ? ps / (fmaxf(nc, 1.f) * (float)D_FEAT) : 0.f;
    }
    cons /= (float)(F_FRAMES - 1);
    float temp = 0.f;
    for (int f = 0; f < F_FRAMES; ++f) {
      temp += acc[f] / (fmaxf(acc[8 + f], 1.f) * (float)D_FEAT);
    }
    temp /= (float)F_FRAMES;
    out[0] = cons;                    // consistency_loss
    out[1] = temp;                    // temporal_smoothness_loss
    out[2] = cons + 0.5f * temp;      // total_loss
  }
}

extern "C" void kernel_launch(void* const* d_in, const int* in_sizes, int n_in,
                              void* d_out, int out_size, void* d_ws, size_t ws_size,
                              hipStream_t stream) {
  (void)in_sizes; (void)n_in; (void)out_size; (void)ws_size;
  const float* feat = (const float*)d_in[0];   // point_features [F,N,D] f32
  // d_in[1] = visibility_matrix (unused by the reference math)
  const int* idx = (const int*)d_in[2];        // point_indices [F,N] i32

  float* wsf  = (float*)d_ws;                  // acc[32] + zrow[128]
  float* acc  = wsf;
  float* zrow = wsf + ACC_SLOTS;
  int*   pos  = (int*)((char*)d_ws + 1024);    // [F][P] int32 (6.4 MB)
  float* out  = (float*)d_out;

  const int posTotal = F_FRAMES * P_SCENE;
  cfc_init<<<(posTotal + 255) / 256, 256, 0, stream>>>(wsf, pos, posTotal);
  cfc_scatter<<<(F_FRAMES * N_PTS + 255) / 256, 256, 0, stream>>>(idx, pos);
  cfc_main<<<2048, 256, 0, stream>>>(feat, pos, zrow, acc);
  cfc_finalize<<<1, 32, 0, stream>>>(acc, out);
}